// ProjectionLoss_23003844837928
// MI455X (gfx1250) — compile-verified
//
#include <hip/hip_runtime.h>

typedef __attribute__((ext_vector_type(2))) float v2f;
typedef __attribute__((ext_vector_type(8))) float v8f;

#define NPIX 4096
#define NB   64
#define NP   1024

// View-1 projection (R = identity): u=(120x+32z+80)/(z+2.5), v=(120y+32z+80)/(z+2.5)
__device__ __forceinline__ int proj_idx(float x, float y, float z) {
  float w = z + 2.5f;
  float u = (120.0f * x + 32.0f * z + 80.0f) / w;
  float v = (120.0f * y + 32.0f * z + 80.0f) / w;
  int idx = (int)(64.0f * rintf(u) + rintf(v));   // rintf = round-half-even, matches jnp.round
  idx = idx < 0 ? 0 : idx;
  idx = idx > 4095 ? 4095 : idx;
  return idx;
}

// One workgroup per batch: signed histogram (hist_gt - hist_res) via LDS float atomics.
__global__ __launch_bounds__(1024) void hist_kernel(
    const float* __restrict__ gt, const float* __restrict__ res,
    float* __restrict__ hist) {
  __shared__ float lh[NPIX];
  const int b = blockIdx.x;
  const int t = threadIdx.x;
  for (int i = t; i < NPIX; i += 1024) lh[i] = 0.0f;
  __syncthreads();
  const size_t base = ((size_t)b * NP + t) * 3;
  float gx = gt[base + 0], gy = gt[base + 1], gz = gt[base + 2];
  float rx = res[base + 0], ry = res[base + 1], rz = res[base + 2];
  atomicAdd(&lh[proj_idx(gx, gy, gz)],  1.0f);
  atomicAdd(&lh[proj_idx(rx, ry, rz)], -1.0f);
  __syncthreads();
  float* hb = hist + (size_t)b * NPIX;
  for (int i = t; i < NPIX; i += 1024) hb[i] = lh[i];
}

// D[64,4096] = H[64,4096] @ plate[4096,4096] with V_WMMA_F32_16X16X4_F32,
// fused with per-wave sum(|D|). One wave per 16-col tile; 4 M-tiles per wave
// so the plate (B) fragment is reused 4x per K-step.
__global__ __launch_bounds__(256) void gemm_abs_kernel(
    const float* __restrict__ hist, const float* __restrict__ plate,
    float* __restrict__ partial) {
  const int lane = threadIdx.x & 31;
  const int wave = (blockIdx.x << 3) | (threadIdx.x >> 5);  // 0..255 => N-tile
  const int n0   = wave << 4;
  const int lm   = lane & 15;        // row (A) / col (B) within tile
  const int kh   = (lane >> 4) << 1; // lanes 16-31 hold K+2 (ISA 7.12.2 f32 layout)

  const float* a0p = hist + (size_t)(lm +  0) * NPIX;
  const float* a1p = hist + (size_t)(lm + 16) * NPIX;
  const float* a2p = hist + (size_t)(lm + 32) * NPIX;
  const float* a3p = hist + (size_t)(lm + 48) * NPIX;
  const float* bp  = plate + n0 + lm;

  v8f c0 = {}, c1 = {}, c2 = {}, c3 = {};
  for (int k0 = 0; k0 < NPIX; k0 += 4) {
    const int ka = k0 + kh;
    v2f bf; bf.x = bp[(size_t)(ka + 0) * NPIX]; bf.y = bp[(size_t)(ka + 1) * NPIX];
    v2f a0; a0.x = a0p[ka]; a0.y = a0p[ka + 1];
    v2f a1; a1.x = a1p[ka]; a1.y = a1p[ka + 1];
    v2f a2; a2.x = a2p[ka]; a2.y = a2p[ka + 1];
    v2f a3; a3.x = a3p[ka]; a3.y = a3p[ka + 1];
    c0 = __builtin_amdgcn_wmma_f32_16x16x4_f32(false, a0, false, bf, (short)0, c0, false, false);
    c1 = __builtin_amdgcn_wmma_f32_16x16x4_f32(false, a1, false, bf, (short)0, c1, false, false);
    c2 = __builtin_amdgcn_wmma_f32_16x16x4_f32(false, a2, false, bf, (short)0, c2, false, false);
    c3 = __builtin_amdgcn_wmma_f32_16x16x4_f32(false, a3, false, bf, (short)0, c3, false, false);
  }

  float s = 0.0f;
#pragma unroll
  for (int i = 0; i < 8; ++i)
    s += fabsf(c0[i]) + fabsf(c1[i]) + fabsf(c2[i]) + fabsf(c3[i]);
#pragma unroll
  for (int off = 16; off > 0; off >>= 1) s += __shfl_xor(s, off, 32);
  if (lane == 0) partial[wave] = s;
}

// Deterministic final reduction: 256 partials -> loss/64.
__global__ __launch_bounds__(256) void reduce_kernel(
    const float* __restrict__ partial, float* __restrict__ out) {
  __shared__ float sm[256];
  const int t = threadIdx.x;
  sm[t] = partial[t];
  __syncthreads();
  for (int off = 128; off > 0; off >>= 1) {
    if (t < off) sm[t] += sm[t + off];
    __syncthreads();
  }
  if (t == 0) out[0] = sm[0] * (1.0f / (float)NB);
}

extern "C" void kernel_launch(void* const* d_in, const int* in_sizes, int n_in,
                              void* d_out, int out_size, void* d_ws, size_t ws_size,
                              hipStream_t stream) {
  const float* gt    = (const float*)d_in[0];  // [64,1024,3]
  const float* res   = (const float*)d_in[1];  // [64,1024,3]
  const float* plate = (const float*)d_in[2];  // [4096,4096]
  float* hist    = (float*)d_ws;               // 64*4096 floats = 1 MB
  float* partial = hist + (size_t)NB * NPIX;   // 256 floats
  float* out     = (float*)d_out;

  hist_kernel<<<NB, 1024, 0, stream>>>(gt, res, hist);
  gemm_abs_kernel<<<32, 256, 0, stream>>>(hist, plate, partial);
  reduce_kernel<<<1, 256, 0, stream>>>(partial, out);
}